// SAModule_46170898432079
// MI455X (gfx1250) — compile-verified
//
#include <hip/hip_runtime.h>
#include <hip/hip_bf16.h>

// ---------------- problem constants ----------------
#define B_    8
#define N_    16384
#define CIN_  64
#define S_    1024
#define K_    32
#define R_    (B_ * S_ * K_)      // 262144 grouped rows
#define RAD2_ 0.04f               // 0.2^2
#define EPS_  1e-5f

typedef __attribute__((ext_vector_type(2))) float v2f;
typedef __attribute__((ext_vector_type(8))) float v8f;

// =====================================================================
// 1) Farthest point sampling: one block per batch, sequential over S steps
// =====================================================================
__global__ __launch_bounds__(256)
void fps_kernel(const float* __restrict__ pos, int* __restrict__ cent,
                float* __restrict__ dist) {
    const int b = blockIdx.x;
    const int t = threadIdx.x;
    const float* P = pos + (size_t)b * N_ * 3;
    float* D = dist + (size_t)b * N_;

    for (int i = t; i < N_; i += 256) D[i] = 1e10f;

    __shared__ int   s_far;
    __shared__ float sc[3];
    __shared__ float rv[256];
    __shared__ int   ri[256];
    if (t == 0) s_far = 0;
    __syncthreads();

    for (int s = 0; s < S_; ++s) {
        const int far = s_far;
        if (t == 0) {
            cent[b * S_ + s] = far;
            sc[0] = P[far * 3 + 0];
            sc[1] = P[far * 3 + 1];
            sc[2] = P[far * 3 + 2];
        }
        __syncthreads();
        const float cx = sc[0], cy = sc[1], cz = sc[2];
        float bv = -1.0f; int bi = 0;
        for (int i = t; i < N_; i += 256) {
            float dx = P[i * 3 + 0] - cx;
            float dy = P[i * 3 + 1] - cy;
            float dz = P[i * 3 + 2] - cz;
            float d  = dx * dx + dy * dy + dz * dz;
            float nd = fminf(D[i], d);
            D[i] = nd;
            if (nd > bv) { bv = nd; bi = i; }
        }
        rv[t] = bv; ri[t] = bi;
        __syncthreads();
        for (int off = 128; off > 0; off >>= 1) {
            if (t < off) {
                float v2 = rv[t + off]; int i2 = ri[t + off];
                if (v2 > rv[t] || (v2 == rv[t] && i2 < ri[t])) { rv[t] = v2; ri[t] = i2; }
            }
            __syncthreads();
        }
        if (t == 0) s_far = ri[0];
        __syncthreads();
    }
}

// =====================================================================
// 2) Ball query (one wave32 per center): K lowest-index in-radius points,
//    padded with the first in-radius index. Also builds geo input rows
//    [center(3) | nbr(3) | 0 0]  (padded to 8 for fp32 WMMA K-stepping).
// =====================================================================
__global__ __launch_bounds__(32)
void ballquery_kernel(const float* __restrict__ pos, const int* __restrict__ cent,
                      int* __restrict__ nbr, float* __restrict__ geo) {
    const int bs   = blockIdx.x;           // b*S + s
    const int b    = bs / S_;
    const int lane = threadIdx.x;
    const float* P = pos + (size_t)b * N_ * 3;

    const int   ci = cent[bs];
    const float cx = P[ci * 3 + 0], cy = P[ci * 3 + 1], cz = P[ci * 3 + 2];

    __shared__ int sidx[K_];
    int count = 0;
    for (int base = 0; base < N_ && count < K_; base += 32) {
        const int i = base + lane;
        float dx = P[i * 3 + 0] - cx;
        float dy = P[i * 3 + 1] - cy;
        float dz = P[i * 3 + 2] - cz;
        bool pred = (dx * dx + dy * dy + dz * dz) <= RAD2_;
        unsigned mm = __builtin_amdgcn_ballot_w32(pred);
        int posn = __popc(mm & ((1u << lane) - 1u));
        if (pred) {
            int slot = count + posn;
            if (slot < K_) sidx[slot] = i;
        }
        count += __popc(mm);
    }
    __syncthreads();

    const int cnt = count < K_ ? count : K_;
    const int nb  = (lane < cnt) ? sidx[lane] : sidx[0];

    const int r = bs * K_ + lane;
    nbr[r] = nb;
    float* G = geo + (size_t)r * 8;
    G[0] = cx; G[1] = cy; G[2] = cz;
    G[3] = P[nb * 3 + 0]; G[4] = P[nb * 3 + 1]; G[5] = P[nb * 3 + 2];
    G[6] = 0.0f; G[7] = 0.0f;
}

// Pad w_geo1 [6,64] -> [8,64] with zero rows so the GEMM needs no bounds checks
__global__ __launch_bounds__(256)
void pad_wgeo_kernel(const float* __restrict__ w, float* __restrict__ wpad) {
    const int i = blockIdx.x * 256 + threadIdx.x;
    if (i < 8 * 64) wpad[i] = (i < 6 * 64) ? w[i] : 0.0f;
}

// =====================================================================
// 3) WMMA fp32 GEMM:  out[R,COUT] = A[R,CINPAD] * W[CINPAD,COUT] + bias
//    One wave per 16x16 output tile; K stepped by 4 via
//    V_WMMA_F32_16X16X4_F32.  Compile-time shapes -> fully unrolled
//    MAC loop, B fragments preloaded in registers, zero branches,
//    EXEC all-ones at every WMMA.
//    GATHER: A rows fetched via nbr indices (grouped-feature gather).
// =====================================================================
template <int CINPAD, int COUT, bool GATHER>
__global__ __launch_bounds__(128)
void gemm_wmma_kernel(const float* __restrict__ A,
                      const float* __restrict__ W,
                      const float* __restrict__ bias,
                      const int* __restrict__ nbr,
                      float* __restrict__ out) {
    const int lane  = threadIdx.x & 31;
    const int wave  = threadIdx.x >> 5;
    const int mtile = blockIdx.x * 4 + wave;
    const int ntile = blockIdx.y;
    const int hi    = lane >> 4;          // 0: K pair {0,1}, 1: K pair {2,3}
    const int ml    = lane & 15;
    const int col   = ntile * 16 + ml;    // N index (B / C / D column)
    const int row   = mtile * 16 + ml;    // M row this lane streams for A

    size_t abase;
    if constexpr (GATHER) {
        const int bb = row >> 15;         // row / (S_*K_)
        abase = ((size_t)bb * N_ + nbr[row]) * (size_t)CINPAD;
    } else {
        abase = (size_t)row * (size_t)CINPAD;
    }

    constexpr int NK = CINPAD / 4;

    // Preload entire B column strip into registers (NK * 2 VGPRs)
    v2f bf[NK];
#pragma unroll
    for (int j = 0; j < NK; ++j) {
        const int k0 = j * 4 + hi * 2;
        bf[j].x = W[(size_t)k0 * COUT + col];
        bf[j].y = W[(size_t)(k0 + 1) * COUT + col];
    }

    v8f acc = {0.f, 0.f, 0.f, 0.f, 0.f, 0.f, 0.f, 0.f};
#pragma unroll
    for (int j = 0; j < NK; ++j) {
        // A fragment: lanes 0-15 hold K={4j,4j+1}, lanes 16-31 K={4j+2,4j+3}
        v2f a = *(const v2f*)(A + abase + j * 4 + hi * 2);
        acc = __builtin_amdgcn_wmma_f32_16x16x4_f32(
            false, a, false, bf[j], (short)0, acc, false, false);
    }

    const float bb = bias[col];
#pragma unroll
    for (int i = 0; i < 8; ++i) {
        const int m = mtile * 16 + i + hi * 8;   // C/D layout: VGPR i -> M = i + 8*hi
        out[(size_t)m * COUT + col] = acc[i] + bb;
    }
}

// =====================================================================
// 4) Deterministic per-channel sum / sum-of-squares (two-stage, no atomics)
// =====================================================================
#define STAT_BLOCKS 512

__global__ __launch_bounds__(256)
void stats_kernel(const float* __restrict__ h, float* __restrict__ partial,
                  int R, int C) {
    __shared__ float ls[256], lq[256];
    const int t    = threadIdx.x;
    const int c    = t % C;               // C divides 256 (64 or 128)
    const int rpt  = 256 / C;
    const int rsub = t / C;
    float s = 0.0f, q = 0.0f;
    for (int r = blockIdx.x * rpt + rsub; r < R; r += gridDim.x * rpt) {
        float v = h[(size_t)r * C + c];
        s += v; q += v * v;
    }
    ls[t] = s; lq[t] = q;
    __syncthreads();
    if (t < C) {
        float st = ls[t], qt = lq[t];
        for (int j = 1; j < rpt; ++j) { st += ls[t + j * C]; qt += lq[t + j * C]; }
        partial[(size_t)blockIdx.x * 2 * C + t]     = st;
        partial[(size_t)blockIdx.x * 2 * C + C + t] = qt;
    }
}

__global__ __launch_bounds__(256)
void stats_reduce_kernel(const float* __restrict__ partial, float* __restrict__ sums,
                         int C) {
    const int t = threadIdx.x;
    if (t < 2 * C) {
        float s = 0.0f;
        for (int j = 0; j < STAT_BLOCKS; ++j) s += partial[(size_t)j * 2 * C + t];
        sums[t] = s;
    }
}

// Fold BN stats into per-channel scale/shift:  y = h*scale + shift, then ReLU
__global__ __launch_bounds__(128)
void finalize_stats_kernel(const float* __restrict__ sums,
                           const float* __restrict__ g, const float* __restrict__ be,
                           float* __restrict__ scale, float* __restrict__ shift,
                           int C, float invR) {
    const int c = threadIdx.x;
    if (c < C) {
        float mu  = sums[c] * invR;
        float var = sums[C + c] * invR - mu * mu;
        float sc  = g[c] * rsqrtf(var + EPS_);
        scale[c] = sc;
        shift[c] = be[c] - mu * sc;
    }
}

// In-place normalize + ReLU (float4 vectorized); C is a power of two
template <int C>
__global__ __launch_bounds__(256)
void norm_relu_kernel(float* __restrict__ h, const float* __restrict__ scale,
                      const float* __restrict__ shift) {
    const size_t base = ((size_t)blockIdx.x * 256 + threadIdx.x) * 4;
    const int c = (int)(base & (size_t)(C - 1));
    float4 v = *(float4*)(h + base);
    v.x = fmaxf(v.x * scale[c]     + shift[c],     0.0f);
    v.y = fmaxf(v.y * scale[c + 1] + shift[c + 1], 0.0f);
    v.z = fmaxf(v.z * scale[c + 2] + shift[c + 2], 0.0f);
    v.w = fmaxf(v.w * scale[c + 3] + shift[c + 3], 0.0f);
    *(float4*)(h + base) = v;
}

// =====================================================================
// 5) Fused normalize + ReLU + max over K neighbors (final layer)
// =====================================================================
__global__ __launch_bounds__(128)
void maxpool_kernel(const float* __restrict__ h, const float* __restrict__ scale,
                    const float* __restrict__ shift, float* __restrict__ out, int C) {
    const int bs = blockIdx.x;
    const int c  = threadIdx.x;
    const float sc = scale[c], sh = shift[c];
    const float* hp = h + (size_t)bs * K_ * C + c;
    float m = 0.0f;                         // ReLU outputs are >= 0
    for (int k = 0; k < K_; ++k) {
        float v = fmaxf(hp[(size_t)k * C] * sc + sh, 0.0f);
        m = fmaxf(m, v);
    }
    out[(size_t)bs * C + c] = m;
}

// =====================================================================
// launcher
// =====================================================================
extern "C" void kernel_launch(void* const* d_in, const int* in_sizes, int n_in,
                              void* d_out, int out_size, void* d_ws, size_t ws_size,
                              hipStream_t stream) {
    const float* pos     = (const float*)d_in[0];
    const float* feat    = (const float*)d_in[1];
    const float* w_sem1  = (const float*)d_in[2];
    const float* b_sem1  = (const float*)d_in[3];
    const float* g_sem1  = (const float*)d_in[4];
    const float* be_sem1 = (const float*)d_in[5];
    const float* w_sem2  = (const float*)d_in[6];
    const float* b_sem2  = (const float*)d_in[7];
    const float* g_sem2  = (const float*)d_in[8];
    const float* be_sem2 = (const float*)d_in[9];
    const float* w_geo1  = (const float*)d_in[10];
    const float* b_geo1  = (const float*)d_in[11];
    const float* g_geo1  = (const float*)d_in[12];
    const float* be_geo1 = (const float*)d_in[13];
    const float* w_geo2  = (const float*)d_in[14];
    const float* b_geo2  = (const float*)d_in[15];
    const float* g_geo2  = (const float*)d_in[16];
    const float* be_geo2 = (const float*)d_in[17];
    float* out = (float*)d_out;

    // ---- workspace layout (bytes) ----
    char* ws = (char*)d_ws;
    int*   centroids = (int*)(ws + 0);                    //  32 KB
    int*   nbr       = (int*)(ws + 32768);                //   1 MB
    float* distb     = (float*)(ws + 32768 + 1048576);    // 512 KB (reused as stats partials)
    float* partial   = distb;                             // STAT_BLOCKS*2*128*4 = 512 KB
    float* sums      = (float*)(ws + 1605632);            // 256 floats
    float* scale1    = (float*)(ws + 1606656);
    float* shift1    = (float*)(ws + 1607168);
    float* scale2    = (float*)(ws + 1607680);
    float* shift2    = (float*)(ws + 1608192);
    float* wpad      = (float*)(ws + 1608704);            // 8*64 floats (padded w_geo1)
    float* geo       = (float*)(ws + 2097152);            // R*8*4   =   8 MB
    float* h1        = (float*)(ws + 2097152 + 8388608);  // R*64*4  =  64 MB (reused by geo1)
    float* h2        = (float*)(ws + 2097152 + 8388608 + 67108864); // R*128*4 = 128 MB

    const float invR = 1.0f / (float)R_;

    // ---- geometry ----
    fps_kernel<<<B_, 256, 0, stream>>>(pos, centroids, distb);
    ballquery_kernel<<<B_ * S_, 32, 0, stream>>>(pos, centroids, nbr, geo);
    pad_wgeo_kernel<<<2, 256, 0, stream>>>(w_geo1, wpad);

    // ---- semantic branch ----
    {   // sem1: gather feat via nbr, 64 -> 64
        dim3 g(R_ / 64, 64 / 16);
        gemm_wmma_kernel<64, 64, true><<<g, 128, 0, stream>>>(feat, w_sem1, b_sem1, nbr, h1);
    }
    stats_kernel<<<STAT_BLOCKS, 256, 0, stream>>>(h1, partial, R_, 64);
    stats_reduce_kernel<<<1, 256, 0, stream>>>(partial, sums, 64);
    finalize_stats_kernel<<<1, 128, 0, stream>>>(sums, g_sem1, be_sem1, scale1, shift1, 64, invR);
    norm_relu_kernel<64><<<(R_ * 64 / 4) / 256, 256, 0, stream>>>(h1, scale1, shift1);
    {   // sem2: 64 -> 128
        dim3 g(R_ / 64, 128 / 16);
        gemm_wmma_kernel<64, 128, false><<<g, 128, 0, stream>>>(h1, w_sem2, b_sem2, nullptr, h2);
    }
    stats_kernel<<<STAT_BLOCKS, 256, 0, stream>>>(h2, partial, R_, 128);
    stats_reduce_kernel<<<1, 256, 0, stream>>>(partial, sums, 128);
    finalize_stats_kernel<<<1, 128, 0, stream>>>(sums, g_sem2, be_sem2, scale2, shift2, 128, invR);
    maxpool_kernel<<<B_ * S_, 128, 0, stream>>>(h2, scale2, shift2, out, 128);

    // ---- geometric branch ----
    {   // geo1: padded Cin 8 -> 64 (wpad has zero rows 6,7)
        dim3 g(R_ / 64, 64 / 16);
        gemm_wmma_kernel<8, 64, false><<<g, 128, 0, stream>>>(geo, wpad, b_geo1, nullptr, h1);
    }
    stats_kernel<<<STAT_BLOCKS, 256, 0, stream>>>(h1, partial, R_, 64);
    stats_reduce_kernel<<<1, 256, 0, stream>>>(partial, sums, 64);
    finalize_stats_kernel<<<1, 128, 0, stream>>>(sums, g_geo1, be_geo1, scale1, shift1, 64, invR);
    norm_relu_kernel<64><<<(R_ * 64 / 4) / 256, 256, 0, stream>>>(h1, scale1, shift1);
    {   // geo2: 64 -> 128
        dim3 g(R_ / 64, 128 / 16);
        gemm_wmma_kernel<64, 128, false><<<g, 128, 0, stream>>>(h1, w_geo2, b_geo2, nullptr, h2);
    }
    stats_kernel<<<STAT_BLOCKS, 256, 0, stream>>>(h2, partial, R_, 128);
    stats_reduce_kernel<<<1, 256, 0, stream>>>(partial, sums, 128);
    finalize_stats_kernel<<<1, 128, 0, stream>>>(sums, g_geo2, be_geo2, scale2, shift2, 128, invR);
    maxpool_kernel<<<B_ * S_, 128, 0, stream>>>(h2, scale2, shift2,
                                                out + (size_t)B_ * S_ * 128, 128);
}